// SAModule_msg_43997644980918
// MI455X (gfx1250) — compile-verified
//
#include <hip/hip_runtime.h>
#include <hip/hip_bf16.h>
#include <stdint.h>

#define NPTS 16384
#define MCTR 4096
#define FIN  64

typedef __attribute__((ext_vector_type(2))) float v2f;
typedef __attribute__((ext_vector_type(8))) float v8f;
typedef __attribute__((ext_vector_type(4))) float vf4;

// ---------------------------------------------------------------------------
// Kernel 0: zero the whole output (x_out needs 0-init for atomic max; batch
// region needs zeros; new_pos overwritten later by FPS).
// ---------------------------------------------------------------------------
__global__ void zero_kernel(float* __restrict__ p, int n) {
  int i = blockIdx.x * blockDim.x + threadIdx.x;
  if (i < n) p[i] = 0.0f;
}

// ---------------------------------------------------------------------------
// Kernel 1: Farthest Point Sampling. One workgroup (1024 thr = 32 waves),
// 16 points per thread held in registers. Matches jnp ops: no fp contraction,
// min update then argmax with first-index tie-break.
// ---------------------------------------------------------------------------
__global__ __launch_bounds__(1024) void fps_kernel(const float* __restrict__ pos,
                                                   int* __restrict__ idx,
                                                   float* __restrict__ newpos) {
#pragma clang fp contract(off)
  const int t    = threadIdx.x;
  const int lane = t & 31;
  const int wid  = t >> 5;
  __shared__ float redv[32];
  __shared__ int   redi[32];
  __shared__ int   bcast;

  float px[16], py[16], pz[16], mind[16];
  for (int q = 0; q < 16; ++q) {
    int p = t + (q << 10);
    px[q] = pos[p * 3 + 0];
    py[q] = pos[p * 3 + 1];
    pz[q] = pos[p * 3 + 2];
    mind[q] = __builtin_inff();
  }

  float lx = pos[0], ly = pos[1], lz = pos[2];
  if (t == 0) {
    idx[0] = 0;
    newpos[0] = lx; newpos[1] = ly; newpos[2] = lz;
  }

  for (int it = 1; it < MCTR; ++it) {
    float bv = -__builtin_inff();
    int   bi = 0x7fffffff;
    for (int q = 0; q < 16; ++q) {
      float dx = px[q] - lx, dy = py[q] - ly, dz = pz[q] - lz;
      float d  = (dx * dx + dy * dy) + dz * dz;
      float m  = fminf(mind[q], d);
      mind[q]  = m;
      if (m > bv) { bv = m; bi = t + (q << 10); }
    }
    for (int m = 16; m >= 1; m >>= 1) {
      float ov = __shfl_xor(bv, m);
      int   oi = __shfl_xor(bi, m);
      if (ov > bv || (ov == bv && oi < bi)) { bv = ov; bi = oi; }
    }
    if (lane == 0) { redv[wid] = bv; redi[wid] = bi; }
    __syncthreads();
    if (wid == 0) {
      bv = redv[lane]; bi = redi[lane];
      for (int m = 16; m >= 1; m >>= 1) {
        float ov = __shfl_xor(bv, m);
        int   oi = __shfl_xor(bi, m);
        if (ov > bv || (ov == bv && oi < bi)) { bv = ov; bi = oi; }
      }
      if (lane == 0) {
        bcast   = bi;
        idx[it] = bi;
        newpos[it * 3 + 0] = pos[bi * 3 + 0];
        newpos[it * 3 + 1] = pos[bi * 3 + 1];
        newpos[it * 3 + 2] = pos[bi * 3 + 2];
      }
    }
    __syncthreads();
    int nb = bcast;
    lx = pos[nb * 3 + 0]; ly = pos[nb * 3 + 1]; lz = pos[nb * 3 + 2];
  }
}

// ---------------------------------------------------------------------------
// Kernel 2: per-center sorted top-64 by squared distance (pruned at r_max^2).
// One wave per center; sorted list kept in LDS; ballot-guarded parallel
// insertion (all 32 lanes cooperate on each shift).
// ---------------------------------------------------------------------------
__global__ __launch_bounds__(256) void knn_kernel(const float* __restrict__ pos,
                                                  const float* __restrict__ newpos,
                                                  int* __restrict__ knn,
                                                  int* __restrict__ cnt0,
                                                  int* __restrict__ cnt1,
                                                  int* __restrict__ cnt2) {
#pragma clang fp contract(off)
  const int lane   = threadIdx.x & 31;
  const int w      = threadIdx.x >> 5;
  const int center = blockIdx.x * 8 + w;

  volatile __shared__ float bd[8][64];
  volatile __shared__ int   bix[8][64];

  const float INF = __builtin_inff();
  bd[w][lane]      = INF; bd[w][lane + 32]  = INF;
  bix[w][lane]     = 0;   bix[w][lane + 32] = 0;
  __builtin_amdgcn_wave_barrier();

  const float rr0 = (float)(0.1 * 0.1);   // match f64 r*r -> f32 cast in jax
  const float rr1 = (float)(0.2 * 0.2);
  const float rr2 = (float)(0.4 * 0.4);

  float cx = newpos[center * 3 + 0];
  float cy = newpos[center * 3 + 1];
  float cz = newpos[center * 3 + 2];
  float c2 = (cx * cx + cy * cy) + cz * cz;

  for (int j0 = 0; j0 < NPTS; j0 += 32) {
    int   j  = j0 + lane;
    float qx = pos[j * 3 + 0], qy = pos[j * 3 + 1], qz = pos[j * 3 + 2];
    float b2 = (qx * qx + qy * qy) + qz * qz;
    float dt = (cx * qx + cy * qy) + cz * qz;
    float d2 = (c2 + b2) - 2.0f * dt;

    bool cand = (d2 <= rr2) && (d2 < bd[w][63]);
    unsigned long long m = __ballot(cand);
    while (m) {
      int src = __ffsll(m) - 1;
      m &= m - 1;
      float dc = __shfl(d2, src);
      int   jc = __shfl(j, src);
      if (dc >= bd[w][63]) continue;
      // insertion position = #entries <= dc (stable: equal keys stay first)
      bool le1 = bd[w][lane]      <= dc;
      bool le2 = bd[w][lane + 32] <= dc;
      int  p   = __popcll(__ballot(le1)) + __popcll(__ballot(le2));
      // parallel shift: lane owns slots {lane, lane+32}
      float cur1 = bd[w][lane];        int ci1 = bix[w][lane];
      float cur2 = bd[w][lane + 32];   int ci2 = bix[w][lane + 32];
      float pr1  = (lane > 0) ? bd[w][lane - 1] : 0.0f;
      int   pi1  = (lane > 0) ? bix[w][lane - 1] : 0;
      float pr2  = bd[w][lane + 31];   int pi2 = bix[w][lane + 31];
      __builtin_amdgcn_wave_barrier();
      int s1 = lane, s2 = lane + 32;
      bd[w][s1]  = (s1 < p) ? cur1 : ((s1 == p) ? dc : pr1);
      bix[w][s1] = (s1 < p) ? ci1  : ((s1 == p) ? jc : pi1);
      bd[w][s2]  = (s2 < p) ? cur2 : ((s2 == p) ? dc : pr2);
      bix[w][s2] = (s2 < p) ? ci2  : ((s2 == p) ? jc : pi2);
      __builtin_amdgcn_wave_barrier();
    }
  }

  bool i0  = (lane < 16) && (bd[w][lane] <= rr0);
  bool i1  = (bd[w][lane] <= rr1);
  bool i2a = (bd[w][lane] <= rr2);
  bool i2b = (bd[w][lane + 32] <= rr2);
  int c0  = __popcll(__ballot(i0));
  int c1  = __popcll(__ballot(i1));
  int c2v = __popcll(__ballot(i2a)) + __popcll(__ballot(i2b));

  knn[center * 64 + lane]      = bix[w][lane];
  knn[center * 64 + lane + 32] = bix[w][lane + 32];
  if (lane == 0) { cnt0[center] = c0; cnt1[center] = c1; cnt2[center] = c2v; }
}

// ---------------------------------------------------------------------------
// Kernel 3: gather + 3-layer MLP (fp32 WMMA 16x16x4) + masked maxpool.
// One wave per 16-row tile (one tile = 16 neighbors of one center).
// ---------------------------------------------------------------------------
#define LSTRIDE 100  // LDS row stride (floats): 16B-aligned, conflict-free

template <int KDIM, int COUT, bool GUARD67>
__device__ __forceinline__ void mlp_layer(const float* in, float* out,
                                          const float* __restrict__ W,
                                          const float* __restrict__ Bv, int lane) {
  const int row  = lane & 15;
  const int koff = (lane >> 4) * 2;
  const int ncol = lane & 15;
  for (int cb = 0; cb < COUT; cb += 16) {
    float bias = Bv[cb + ncol];
    v8f acc;
    for (int g = 0; g < 8; ++g) acc[g] = bias;
    for (int kc = 0; kc < (KDIM + 3) / 4; ++kc) {
      int k0 = kc * 4 + koff;
      v2f a = *(const v2f*)(in + row * LSTRIDE + k0);
      v2f b;
      if (GUARD67) {
        b.x = (k0     < 67) ? W[(k0    ) * COUT + cb + ncol] : 0.0f;
        b.y = (k0 + 1 < 67) ? W[(k0 + 1) * COUT + cb + ncol] : 0.0f;
      } else {
        b.x = W[(k0    ) * COUT + cb + ncol];
        b.y = W[(k0 + 1) * COUT + cb + ncol];
      }
      acc = __builtin_amdgcn_wmma_f32_16x16x4_f32(false, a, false, b,
                                                  (short)0, acc, false, false);
    }
    for (int g = 0; g < 8; ++g) {
      float v = fmaxf(acc[g], 0.0f);             // ReLU
      int orow = (lane < 16) ? g : (g + 8);
      out[orow * LSTRIDE + cb + ncol] = v;
    }
  }
}

template <int KDIM, int COUT>
__device__ __forceinline__ void mlp_final(const float* in,
                                          const float* __restrict__ W,
                                          const float* __restrict__ Bv, int lane,
                                          int jbase, int cnt,
                                          unsigned int* __restrict__ outrow) {
  const int row  = lane & 15;
  const int koff = (lane >> 4) * 2;
  const int ncol = lane & 15;
  for (int cb = 0; cb < COUT; cb += 16) {
    float bias = Bv[cb + ncol];
    v8f acc;
    for (int g = 0; g < 8; ++g) acc[g] = bias;
    for (int kc = 0; kc < (KDIM + 3) / 4; ++kc) {
      int k0 = kc * 4 + koff;
      v2f a = *(const v2f*)(in + row * LSTRIDE + k0);
      v2f b;
      b.x = W[(k0    ) * COUT + cb + ncol];
      b.y = W[(k0 + 1) * COUT + cb + ncol];
      acc = __builtin_amdgcn_wmma_f32_16x16x4_f32(false, a, false, b,
                                                  (short)0, acc, false, false);
    }
    // masked maxpool over this tile's 16 rows (all values >= 0 post-ReLU,
    // so invalid rows contribute 0 harmlessly; output is 0-initialized)
    float mx = 0.0f;
    for (int g = 0; g < 8; ++g) {
      float v = fmaxf(acc[g], 0.0f);
      int r = (lane < 16) ? g : (g + 8);
      mx = fmaxf(mx, ((jbase + r) < cnt) ? v : 0.0f);
    }
    mx = fmaxf(mx, __shfl_xor(mx, 16));
    if (lane < 16)
      atomicMax(outrow + cb + ncol, __float_as_uint(mx));  // monotone for >=0
  }
}

template <int KNB, int C1, int C2, int C3>
__global__ __launch_bounds__(128) void mlp_kernel(
    const float* __restrict__ x, const float* __restrict__ pos,
    const float* __restrict__ newpos, const int* __restrict__ knn,
    const int* __restrict__ cnt,
    const float* __restrict__ W1, const float* __restrict__ B1,
    const float* __restrict__ W2, const float* __restrict__ B2,
    const float* __restrict__ W3, const float* __restrict__ B3,
    unsigned int* __restrict__ outbase) {
  __shared__ float bufA[4][16 * LSTRIDE];
  __shared__ float bufB[4][16 * LSTRIDE];
  const int lane   = threadIdx.x & 31;
  const int w      = threadIdx.x >> 5;
  const int tile   = blockIdx.x * 4 + w;
  const int center = (tile * 16) / KNB;
  const int jbase  = (tile * 16) % KNB;
  const int row    = lane & 15;
  const int hseg   = lane >> 4;
  float* A  = &bufA[w][0];
  float* Bf = &bufB[w][0];

  // gather: row = [ x[col] (64) | pos[col]-newpos[center] (3) | 0 pad ]
  {
    int col = knn[center * 64 + jbase + row];
    const float* xr = x + (size_t)col * FIN;
    if (hseg == 0) {
      for (int c = 0; c < 8; ++c)
        *(vf4*)(A + row * LSTRIDE + c * 4) = *(const vf4*)(xr + c * 4);
      vf4 rv;
      rv.x = pos[col * 3 + 0] - newpos[center * 3 + 0];
      rv.y = pos[col * 3 + 1] - newpos[center * 3 + 1];
      rv.z = pos[col * 3 + 2] - newpos[center * 3 + 2];
      rv.w = 0.0f;
      *(vf4*)(A + row * LSTRIDE + 64) = rv;
    } else {
      for (int c = 8; c < 16; ++c)
        *(vf4*)(A + row * LSTRIDE + c * 4) = *(const vf4*)(xr + c * 4);
    }
  }
  __syncthreads();

  mlp_layer<68, C1, true >(A,  Bf, W1, B1, lane);   // 67 real + 1 zero pad
  __syncthreads();
  mlp_layer<C1, C2, false>(Bf, A,  W2, B2, lane);
  __syncthreads();
  int cv = cnt[center];
  mlp_final<C2, C3>(A, W3, B3, lane, jbase, cv, outbase + (size_t)center * 320);
}

// ---------------------------------------------------------------------------
// Host launcher
// ---------------------------------------------------------------------------
extern "C" void kernel_launch(void* const* d_in, const int* in_sizes, int n_in,
                              void* d_out, int out_size, void* d_ws, size_t ws_size,
                              hipStream_t stream) {
  const float* x   = (const float*)d_in[0];
  const float* pos = (const float*)d_in[1];
  // d_in[2] = batch (all zeros; output batch region zero-filled below)
  const float* W[3][3];
  const float* B[3][3];
  int p = 3;
  for (int s = 0; s < 3; ++s)
    for (int l = 0; l < 3; ++l) {
      W[s][l] = (const float*)d_in[p++];
      B[s][l] = (const float*)d_in[p++];
    }

  float* out    = (float*)d_out;
  float* newpos = out + (size_t)MCTR * 320;  // [M,3] after x_out

  char* ws  = (char*)d_ws;
  int* idx  = (int*)ws;                                  // M
  int* knn  = (int*)(ws + MCTR * 4);                     // M*64
  int* cnt0 = (int*)(ws + MCTR * 4 + MCTR * 64 * 4);
  int* cnt1 = cnt0 + MCTR;
  int* cnt2 = cnt1 + MCTR;

  zero_kernel<<<(out_size + 255) / 256, 256, 0, stream>>>(out, out_size);
  fps_kernel<<<1, 1024, 0, stream>>>(pos, idx, newpos);
  knn_kernel<<<MCTR / 8, 256, 0, stream>>>(pos, newpos, knn, cnt0, cnt1, cnt2);

  unsigned int* uout = (unsigned int*)out;
  mlp_kernel<16, 32, 32, 64><<<1024, 128, 0, stream>>>(
      x, pos, newpos, knn, cnt0,
      W[0][0], B[0][0], W[0][1], B[0][1], W[0][2], B[0][2], uout + 0);
  mlp_kernel<32, 64, 64, 128><<<2048, 128, 0, stream>>>(
      x, pos, newpos, knn, cnt1,
      W[1][0], B[1][0], W[1][1], B[1][1], W[1][2], B[1][2], uout + 64);
  mlp_kernel<64, 64, 96, 128><<<4096, 128, 0, stream>>>(
      x, pos, newpos, knn, cnt2,
      W[2][0], B[2][0], W[2][1], B[2][1], W[2][2], B[2][2], uout + 192);
}